// Tree_49718541418874
// MI455X (gfx1250) — compile-verified
//
#include <hip/hip_runtime.h>
#include <hip/hip_bf16.h>

#define LEAKY_S  0.01f
#define BN_EPS_S 1e-5f
#define OUTW     2048
#define LSTRIDE  1040   // 1040 % 64 == 16 -> adjacent batch rows use disjoint LDS bank groups

// workspace layout (floats):
//  [0..15]  sum of leaky(w1*x+b1) per feature     [16..31] sumsq
//  [32]     sum of t                               [33]     sumsq of t
//  [64 .. 64+B)                t[b]
//  [P1 .. P1+64*32)            stage-1 per-block partials
//  [P2 .. P2+128*2)            stage-2 per-block partials
#define WS_T   64
#define WS_P1  33024
#define WS_P2  35072

typedef float v2f __attribute__((ext_vector_type(2)));
typedef float v8f __attribute__((ext_vector_type(8)));

__device__ __forceinline__ float leaky_f(float v) { return v > 0.0f ? v : LEAKY_S * v; }

// fast sigmoid: v_exp_f32 + v_rcp_f32 (avoids the IEEE v_div_scale/div_fixup chain;
// ~1 ulp rcp error is irrelevant vs validation tolerance, and keeps the 33.5M
// sigmoids of this dispatch far below the 256MB-store bandwidth floor)
__device__ __forceinline__ float fast_sigmoid(float z) {
    return __builtin_amdgcn_rcpf(1.0f + __expf(-z));
}

__device__ __forceinline__ float wave_sum(float v) {
#pragma unroll
    for (int off = 16; off > 0; off >>= 1) v += __shfl_xor(v, off, 32);
    return v;
}

// ---- pass 1: per-feature sum/sumsq of leaky(w1[j]*x + b1[j]) over the batch ----
__global__ void k_stats1(const float* __restrict__ x, const float* __restrict__ w1,
                         const float* __restrict__ b1, float* __restrict__ ws, int n) {
    __shared__ float red[8][32];
    float s[16], q[16];
#pragma unroll
    for (int j = 0; j < 16; ++j) { s[j] = 0.0f; q[j] = 0.0f; }
    for (int i = blockIdx.x * blockDim.x + threadIdx.x; i < n; i += gridDim.x * blockDim.x) {
        float xv = x[i];
#pragma unroll
        for (int j = 0; j < 16; ++j) {
            float h = leaky_f(fmaf(w1[j], xv, b1[j]));
            s[j] += h; q[j] += h * h;
        }
    }
    int lane = threadIdx.x & 31, wv = threadIdx.x >> 5;
#pragma unroll
    for (int j = 0; j < 16; ++j) {
        float ss = wave_sum(s[j]);
        float qq = wave_sum(q[j]);
        if (lane == 0) { red[wv][j] = ss; red[wv][16 + j] = qq; }
    }
    __syncthreads();
    if (threadIdx.x < 32) {
        float acc = 0.0f;
#pragma unroll
        for (int w = 0; w < 8; ++w) acc += red[w][threadIdx.x];   // fixed order: deterministic
        ws[WS_P1 + blockIdx.x * 32 + threadIdx.x] = acc;
    }
}

__global__ void k_stats1b(float* __restrict__ ws) {
    if (threadIdx.x < 32) {
        float acc = 0.0f;
        for (int b = 0; b < 64; ++b) acc += ws[WS_P1 + b * 32 + threadIdx.x];
        ws[threadIdx.x] = acc;
    }
}

// ---- pass 2: t[b] = leaky( BN1(leaky(w1*x+b1)) . w2 + b2 ), plus sum/sumsq of t ----
__global__ void k_mid(const float* __restrict__ x,
                      const float* __restrict__ w1, const float* __restrict__ b1,
                      const float* __restrict__ g1, const float* __restrict__ be1,
                      const float* __restrict__ w2, const float* __restrict__ b2,
                      float* __restrict__ ws, int n) {
    __shared__ float red[2][8];
    int i = blockIdx.x * blockDim.x + threadIdx.x;
    float invN = 1.0f / (float)n;
    float t = 0.0f;
    if (i < n) {
        float xv = x[i];
        float acc = b2[0];
#pragma unroll
        for (int j = 0; j < 16; ++j) {
            float m  = ws[j] * invN;
            float v  = ws[16 + j] * invN - m * m;
            float h  = leaky_f(fmaf(w1[j], xv, b1[j]));
            float hn = fmaf((h - m) * rsqrtf(v + BN_EPS_S), g1[j], be1[j]);
            acc = fmaf(hn, w2[j], acc);
        }
        t = leaky_f(acc);
        ws[WS_T + i] = t;
    }
    int lane = threadIdx.x & 31, wv = threadIdx.x >> 5;
    float ts = wave_sum(t);
    float qs = wave_sum(t * t);
    if (lane == 0) { red[0][wv] = ts; red[1][wv] = qs; }
    __syncthreads();
    if (threadIdx.x == 0) {
        float a0 = 0.0f, a1 = 0.0f;
#pragma unroll
        for (int w = 0; w < 8; ++w) { a0 += red[0][w]; a1 += red[1][w]; }
        ws[WS_P2 + blockIdx.x * 2 + 0] = a0;
        ws[WS_P2 + blockIdx.x * 2 + 1] = a1;
    }
}

__global__ void k_midb(float* __restrict__ ws) {
    if (threadIdx.x < 2) {
        float acc = 0.0f;
        for (int b = 0; b < 128; ++b) acc += ws[WS_P2 + b * 2 + threadIdx.x];
        ws[32 + threadIdx.x] = acc;
    }
}

// ---- pass 3: WMMA decision logits + soft decision tree, streamed to 256MB output ----
__global__ void __launch_bounds__(256)
k_tree(const float* __restrict__ ws,
       const float* __restrict__ fcw, const float* __restrict__ fcb,
       const float* __restrict__ g2, const float* __restrict__ be2,
       float* __restrict__ out, int n) {
    extern __shared__ float smem[];
    float* dls = smem;                  // [16][LSTRIDE] sigmoid decisions
    float* mul = smem + 16 * LSTRIDE;   // [16][LSTRIDE] mu for internal nodes
    __shared__ float sxn[16];

    const int tid   = threadIdx.x;
    const int bbase = blockIdx.x * 16;

    // BN2-normalized scalar xn for the block's 16 batch rows
    if (tid < 16) {
        float invN = 1.0f / (float)n;
        float mt = ws[32] * invN;
        float vt = ws[33] * invN - mt * mt;
        float t  = ws[WS_T + bbase + tid];
        sxn[tid] = fmaf((t - mt) * rsqrtf(vt + BN_EPS_S), g2[0], be2[0]);
    }
    __syncthreads();

    // phase 1: D = A(16x4: xn in K=0) x B(4x16: fc_w in K=0) + C(fc_b bcast)  (exact rank-1 f32)
    const int lane = tid & 31;
    const int wv   = tid >> 5;
    const int nlo  = lane & 15;
    const int mrow = (lane >> 4) * 8;          // C/D layout: lanes 0-15 -> M=r, lanes 16-31 -> M=8+r
    v2f a;
    a.x = (lane < 16) ? sxn[nlo] : 0.0f;       // A layout: lanes 0-15, VGPR0 = K=0 column
    a.y = 0.0f;
#pragma unroll
    for (int i = 0; i < 8; ++i) {
        int nb = (wv + 8 * i) * 16;            // 8 waves x 8 tiles = 64 node tiles = 1024 nodes
        v2f b;
        b.x = (lane < 16) ? fcw[nb + nlo] : 0.0f;   // B: K=0 row striped over lanes 0-15
        b.y = 0.0f;
        float bias = fcb[nb + nlo];
        v8f c;
#pragma unroll
        for (int r = 0; r < 8; ++r) c[r] = bias;    // C[m,n] = fc_b[n]: constant per lane
        v8f d = __builtin_amdgcn_wmma_f32_16x16x4_f32(false, a, false, b,
                                                      (short)0, c, false, false);
#pragma unroll
        for (int r = 0; r < 8; ++r) {
            dls[(mrow + r) * LSTRIDE + nb + nlo] = fast_sigmoid(d[r]);
        }
    }
    __syncthreads();

    // phase 2: BFS tree routing; 16 sub-lanes per batch row, contiguous float2 child stores
    const int b   = tid >> 4;
    const int sub = tid & 15;
    float* dRow = dls + b * LSTRIDE;
    float* mRow = mul + b * LSTRIDE;
    float* oRow = out + (size_t)(bbase + b) * OUTW;
    if (sub == 0) { oRow[0] = 0.0f; oRow[1] = 1.0f; }
    for (int L = 0; L < 10; ++L) {
        int cnt = 1 << L;
        for (int j = sub; j < cnt; j += 16) {
            int nidx = cnt + j;
            float pm = (L == 0) ? 1.0f : mRow[nidx];
            float dv = dRow[nidx];
            float c0 = pm * dv;
            float c1 = pm - c0;
            if (L < 9) { mRow[2 * nidx] = c0; mRow[2 * nidx + 1] = c1; }
            float2 o; o.x = c0; o.y = c1;
            *reinterpret_cast<float2*>(oRow + 2 * nidx) = o;   // 8B-aligned global_store_b64
        }
        __syncthreads();
    }
}

extern "C" void kernel_launch(void* const* d_in, const int* in_sizes, int n_in,
                              void* d_out, int out_size, void* d_ws, size_t ws_size,
                              hipStream_t stream) {
    const float* x   = (const float*)d_in[0];
    const float* w1  = (const float*)d_in[1];
    const float* b1  = (const float*)d_in[2];
    const float* g1  = (const float*)d_in[3];
    const float* be1 = (const float*)d_in[4];
    const float* w2  = (const float*)d_in[5];
    const float* b2  = (const float*)d_in[6];
    const float* g2  = (const float*)d_in[7];
    const float* be2 = (const float*)d_in[8];
    const float* fcw = (const float*)d_in[9];
    const float* fcb = (const float*)d_in[10];
    float* out = (float*)d_out;
    float* ws  = (float*)d_ws;
    const int n = in_sizes[0];   // 32768

    const size_t ldsBytes = (size_t)2 * 16 * LSTRIDE * sizeof(float);  // ~130 KB < 320 KB/WGP
    (void)hipFuncSetAttribute((const void*)k_tree,
                              hipFuncAttributeMaxDynamicSharedMemorySize, (int)ldsBytes);

    k_stats1 <<<64, 256, 0, stream>>>(x, w1, b1, ws, n);
    k_stats1b<<<1, 32, 0, stream>>>(ws);
    k_mid    <<<(n + 255) / 256, 256, 0, stream>>>(x, w1, b1, g1, be1, w2, b2, ws, n);
    k_midb   <<<1, 64, 0, stream>>>(ws);
    k_tree   <<<n / 16, 256, ldsBytes, stream>>>(ws, fcw, fcb, g2, be2, out, n);
}